// KSparse_738734375123
// MI455X (gfx1250) — compile-verified
//
#include <hip/hip_runtime.h>
#include <hip/hip_bf16.h>

// K-sparse (train branch): per row, threshold = value at descending rank k,
// out = x * (x > threshold). Implemented as exact MSB-first radix select on
// order-preserving unsigned keys; one workgroup (256 threads, 8 wave32) per row.

static constexpr int kThreads = 256;
static constexpr int kN       = 8192;            // feature dim (fixed by reference)
static constexpr int kVPT     = kN / kThreads;   // 32 values per thread, held in VGPRs
static constexpr int kBins0   = 4096;            // pass 0: 12-bit digit
static constexpr int kBins1   = 1024;            // pass 1/2: 10-bit digit

__device__ __forceinline__ unsigned key_of(float f) {
  unsigned b = __float_as_uint(f);
  // monotonic: key order == float order (negatives flipped below positives)
  return (b & 0x80000000u) ? ~b : (b | 0x80000000u);
}
__device__ __forceinline__ float val_of(unsigned k) {
  unsigned b = (k & 0x80000000u) ? (k ^ 0x80000000u) : ~k;
  return __uint_as_float(b);
}

// Given hist[CHUNK*256] (counts within current candidate set) and target
// descending rank krem, find the unique digit d with
//   suffix(d) > krem >= suffix(d+1)      (suffix = inclusive count from top)
// and publish {d, krem - suffix(d+1)} via bcast[0..1].
template <int CHUNK>
__device__ __forceinline__ void select_digit(const unsigned* hist, unsigned* scanbuf,
                                             unsigned* bcast, unsigned krem) {
  const int t = threadIdx.x;
  unsigned partial = 0u;
#pragma unroll
  for (int i = 0; i < CHUNK; ++i) partial += hist[t * CHUNK + i];
  scanbuf[t] = partial;
  __syncthreads();
  // inclusive suffix scan of 256 chunk totals (Hillis-Steele, 8 steps)
#pragma unroll
  for (int off = 1; off < kThreads; off <<= 1) {
    unsigned mine = scanbuf[t];
    unsigned nxt  = (t + off < kThreads) ? scanbuf[t + off] : 0u;
    __syncthreads();
    scanbuf[t] = mine + nxt;
    __syncthreads();
  }
  // walk own chunk from its top digit downward; 'above' = suffix(bin+1)
  unsigned above = (t + 1 < kThreads) ? scanbuf[t + 1] : 0u;
#pragma unroll
  for (int i = CHUNK - 1; i >= 0; --i) {
    unsigned c    = hist[t * CHUNK + i];
    unsigned incl = above + c;
    if (incl > krem && above <= krem) {   // exactly one (bin) satisfies this
      bcast[0] = (unsigned)(t * CHUNK + i);
      bcast[1] = krem - above;
    }
    above = incl;
  }
  __syncthreads();
}

__global__ __launch_bounds__(kThreads)
void ksparse_topk_kernel(const float* __restrict__ x,
                         const int* __restrict__ kptr,
                         float* __restrict__ out) {
  __shared__ unsigned hist[kBins0];      // 16 KB (CDNA5: 320 KB/WGP available)
  __shared__ unsigned scanbuf[kThreads];
  __shared__ unsigned bcast[2];

  const int t = threadIdx.x;
  const size_t rowOff = (size_t)blockIdx.x * (size_t)kN;
  const float4* __restrict__ xr = (const float4*)(x + rowOff);

  // ---- single streaming read of the row: 8 x b128 per thread, coalesced ----
  unsigned keys[kVPT];
#pragma unroll
  for (int j = 0; j < kVPT / 4; ++j) {
    float4 v = xr[j * kThreads + t];
    keys[j * 4 + 0] = key_of(v.x);
    keys[j * 4 + 1] = key_of(v.y);
    keys[j * 4 + 2] = key_of(v.z);
    keys[j * 4 + 3] = key_of(v.w);
  }

  unsigned krem = (unsigned)(*kptr);   // target descending rank (k)
  unsigned pref = 0;

  // ================= pass 0: 12-bit MSB digit, 4096 bins =================
#pragma unroll
  for (int i = 0; i < kBins0 / kThreads; ++i) hist[t + i * kThreads] = 0u;
  __syncthreads();
#pragma unroll
  for (int i = 0; i < kVPT; ++i)
    atomicAdd(&hist[keys[i] >> 20], 1u);           // ds_add_u32
  __syncthreads();
  select_digit<kBins0 / kThreads>(hist, scanbuf, bcast, krem);
  pref = bcast[0];
  krem = bcast[1];

  // ================= pass 1: next 10 bits, 1024 bins ======================
#pragma unroll
  for (int i = 0; i < kBins1 / kThreads; ++i) hist[t + i * kThreads] = 0u;
  __syncthreads();
#pragma unroll
  for (int i = 0; i < kVPT; ++i)
    if ((keys[i] >> 20) == pref)
      atomicAdd(&hist[(keys[i] >> 10) & 1023u], 1u);
  __syncthreads();
  select_digit<kBins1 / kThreads>(hist, scanbuf, bcast, krem);
  pref = (pref << 10) | bcast[0];
  krem = bcast[1];

  // ================= pass 2: low 10 bits, 1024 bins =======================
#pragma unroll
  for (int i = 0; i < kBins1 / kThreads; ++i) hist[t + i * kThreads] = 0u;
  __syncthreads();
#pragma unroll
  for (int i = 0; i < kVPT; ++i)
    if ((keys[i] >> 10) == pref)
      atomicAdd(&hist[keys[i] & 1023u], 1u);
  __syncthreads();
  select_digit<kBins1 / kThreads>(hist, scanbuf, bcast, krem);
  const unsigned tkey = (pref << 10) | bcast[0];   // exact key of sort(x)[N-1-k]

  // ---- single streaming write: emulate x * (x > kth) exactly (keeps -0.0) ----
  float4* __restrict__ orow = (float4*)(out + rowOff);
#pragma unroll
  for (int j = 0; j < kVPT / 4; ++j) {
    float4 o;
    {
      unsigned kk = keys[j * 4 + 0];
      o.x = val_of(kk) * ((kk > tkey) ? 1.0f : 0.0f);
    }
    {
      unsigned kk = keys[j * 4 + 1];
      o.y = val_of(kk) * ((kk > tkey) ? 1.0f : 0.0f);
    }
    {
      unsigned kk = keys[j * 4 + 2];
      o.z = val_of(kk) * ((kk > tkey) ? 1.0f : 0.0f);
    }
    {
      unsigned kk = keys[j * 4 + 3];
      o.w = val_of(kk) * ((kk > tkey) ? 1.0f : 0.0f);
    }
    orow[j * kThreads + t] = o;
  }
}

extern "C" void kernel_launch(void* const* d_in, const int* in_sizes, int n_in,
                              void* d_out, int out_size, void* d_ws, size_t ws_size,
                              hipStream_t stream) {
  const float* x  = (const float*)d_in[0];
  const int*   kp = (const int*)d_in[1];
  float*       o  = (float*)d_out;
  const int rows  = in_sizes[0] / kN;   // B = 4096 for the reference shapes
  ksparse_topk_kernel<<<rows, kThreads, 0, stream>>>(x, kp, o);
}